// CANModule_31301721654020
// MI455X (gfx1250) — compile-verified
//
#include <hip/hip_runtime.h>
#include <hip/hip_bf16.h>

// CANModule on MI455X (gfx1250).
// out[b][h] = relu(place[h] + sum_k ec[b][k]*W_ec[k][h] + sum_i sum_j grid_i[j]*W_mec_i[j][h])
// Pure HBM-bandwidth problem (~369 MB weights, ~0.38 GFLOP). Strategy:
//   - one concatenated-K GEMM, grid_i broadcast to all 4 batch rows (rowStride=0)
//   - V_WMMA_F32_16X16X4_F32 per 4x16 weight tile (fp32 end to end)
//   - K split into 11 chunks of 1024 -> 5632 waves; fp32 atomics into workspace
//   - finalize kernel: + place, ReLU

#define HPC 8192
#define B_ROWS 4
#define KCHUNK 1024

typedef __attribute__((ext_vector_type(2))) float v2f;
typedef __attribute__((ext_vector_type(8))) float v8f;

struct SegTable {
    const float* W[4];     // [K_seg, 8192] weight blocks
    const float* act[4];   // activation rows for each segment
    int rowStride[4];      // 4096 for ec (4 distinct rows), 0 for grids (broadcast)
};

__global__ void __launch_bounds__(256) can_zero_ws(float* __restrict__ p, int n) {
    int i = blockIdx.x * blockDim.x + threadIdx.x;
    if (i < n) p[i] = 0.0f;
}

// grid: (128, 11), block: 128 (4 waves). Wave w of block x owns columns
// h0 = (x*4 + w)*16. blockIdx.y selects (segment, k-chunk).
__global__ void __launch_bounds__(128)
can_wmma_partial(SegTable segs, float* __restrict__ accbuf /* [4][HPC] */) {
    const int lane = threadIdx.x & 31;
    const int wave = threadIdx.x >> 5;
    const int h0   = (blockIdx.x * 4 + wave) * 16;

    // Map chunk index -> (segment, k offset inside segment).
    // K layout: W_ec 4096 (chunks 0-3), W_mec0 1024 (4), W_mec1 2048 (5-6), W_mec2 4096 (7-10)
    const int c = blockIdx.y;
    int seg, klo;
    if (c < 4)      { seg = 0; klo = c * KCHUNK; }
    else if (c < 5) { seg = 1; klo = 0; }
    else if (c < 7) { seg = 2; klo = (c - 5) * KCHUNK; }
    else            { seg = 3; klo = (c - 7) * KCHUNK; }

    const float* __restrict__ W = segs.W[seg];
    const float* __restrict__ A = segs.act[seg];
    const int rs = segs.rowStride[seg];

    // Fragment lane roles (wave32):
    //  A (16x4 f32, 2 VGPRs): lane holds row M = lane&15; VGPR0 = K=koff, VGPR1 = K=koff+1,
    //      koff = 0 for lanes 0-15, 2 for lanes 16-31.
    //  B (4x16 f32, 2 VGPRs): lane holds col N = lane&15; same K striping as A.
    //  D (16x16 f32, 8 VGPRs): lanes 0-15 hold rows 0-7 in d[0..7] -> rows 0-3 = batch rows.
    const int m    = lane & 15;
    const int koff = (lane >> 4) << 1;          // 0 or 2
    const bool arow  = (m < B_ROWS);
    const int  mrow  = arow ? m : 0;            // clamp: keep loads in-bounds
    const float amask = arow ? 1.0f : 0.0f;     // zero rows 4..15 (branch-free, EXEC stays all-1)

    const float* __restrict__ wptr = W + (size_t)(klo + koff) * HPC + h0 + m;
    const float* __restrict__ aptr = A + (size_t)mrow * rs + klo + koff;

    float r0, r1, r2, r3;
#if defined(__has_builtin) && __has_builtin(__builtin_amdgcn_wmma_f32_16x16x4_f32)
    v8f d = {};
#pragma unroll 4
    for (int k = 0; k < KCHUNK; k += 4) {
        v2f a, b;
        a.x = aptr[0] * amask;                  // A[M][k+koff]
        a.y = aptr[1] * amask;                  // A[M][k+koff+1]
        b.x = wptr[0];                          // W[k+koff][h0+N]
        b.y = wptr[HPC];                        // W[k+koff+1][h0+N]
        d = __builtin_amdgcn_wmma_f32_16x16x4_f32(
                /*neg_a=*/false, a, /*neg_b=*/false, b,
                /*c_mod=*/(short)0, d, /*reuse_a=*/false, /*reuse_b=*/false);
        aptr += 4;
        wptr += 4 * HPC;
    }
    r0 = d[0]; r1 = d[1]; r2 = d[2]; r3 = d[3];
#else
    // VALU fallback (host pass / builtin unavailable): same math, no WMMA.
    float acc[B_ROWS] = {0.f, 0.f, 0.f, 0.f};
    if (lane < 16) {
        const float* __restrict__ wp = W + (size_t)klo * HPC + h0 + lane;
        for (int k = 0; k < KCHUNK; ++k) {
            float w = wp[0];
            wp += HPC;
#pragma unroll
            for (int b = 0; b < B_ROWS; ++b)
                acc[b] += A[(size_t)b * rs + klo + k] * w;
        }
    }
    r0 = acc[0]; r1 = acc[1]; r2 = acc[2]; r3 = acc[3];
#endif

    if (lane < 16) {
        const int h = h0 + lane;
        atomicAdd(&accbuf[0 * HPC + h], r0);
        atomicAdd(&accbuf[1 * HPC + h], r1);
        atomicAdd(&accbuf[2 * HPC + h], r2);
        atomicAdd(&accbuf[3 * HPC + h], r3);
    }
}

__global__ void __launch_bounds__(256)
can_finalize(const float* __restrict__ accbuf, const float* __restrict__ place,
             float* __restrict__ out, int n) {
    int i = blockIdx.x * blockDim.x + threadIdx.x;
    if (i < n) {
        float v = place[i & (HPC - 1)] + accbuf[i];
        out[i] = v > 0.0f ? v : 0.0f;
    }
}

extern "C" void kernel_launch(void* const* d_in, const int* in_sizes, int n_in,
                              void* d_out, int out_size, void* d_ws, size_t ws_size,
                              hipStream_t stream) {
    const float* ec    = (const float*)d_in[0];   // (4, 4096)
    const float* place = (const float*)d_in[1];   // (1, 8192)
    const float* g0    = (const float*)d_in[2];   // (1, 1024)
    const float* g1    = (const float*)d_in[3];   // (1, 2048)
    const float* g2    = (const float*)d_in[4];   // (1, 4096)
    const float* W_ec  = (const float*)d_in[5];   // (4096, 8192)
    const float* W_m0  = (const float*)d_in[6];   // (1024, 8192)
    const float* W_m1  = (const float*)d_in[7];   // (2048, 8192)
    const float* W_m2  = (const float*)d_in[8];   // (4096, 8192)
    // d_in[9..14] (W_gh*, W_gg*) are dead: the reference discards new_grids.

    float* accbuf = (float*)d_ws;                 // 4*8192 floats = 128 KB
    const int nacc = B_ROWS * HPC;

    can_zero_ws<<<(nacc + 255) / 256, 256, 0, stream>>>(accbuf, nacc);

    SegTable st;
    st.W[0] = W_ec;  st.act[0] = ec;  st.rowStride[0] = 4096;
    st.W[1] = W_m0;  st.act[1] = g0;  st.rowStride[1] = 0;
    st.W[2] = W_m1;  st.act[2] = g1;  st.rowStride[2] = 0;
    st.W[3] = W_m2;  st.act[3] = g2;  st.rowStride[3] = 0;

    dim3 grid(HPC / 64, 11);   // 128 column-tile blocks x 11 K-chunks
    can_wmma_partial<<<grid, 128, 0, stream>>>(st, accbuf);

    can_finalize<<<(nacc + 255) / 256, 256, 0, stream>>>(accbuf, place, (float*)d_out, nacc);
}